// SHMoofModel_39711267619066
// MI455X (gfx1250) — compile-verified
//
#include <hip/hip_runtime.h>
#include <math.h>

#define MAX_RESOLVE 16
#define TBL_MAX     2048   // merged table capacity (floats); actual n_kmers = 1664
#define BLOCK       256    // 8 waves (wave32)
#define VPT         16     // elements per thread => 4096 per block

// ---------------------------------------------------------------------------
// Phase 1: fold the 16-wide resolve dot-product into a per-kmer scalar.
//   merged[k] = sum_r resolve_weights[k,r] * kmer_embedding[resolve_map[k,r]]
// Tiny: n_kmers (=1664) threads total.
// ---------------------------------------------------------------------------
__global__ void build_merged_kernel(const int*   __restrict__ resolve_map,
                                    const float* __restrict__ resolve_weights,
                                    const float* __restrict__ kmer_embedding,
                                    float*       __restrict__ merged,
                                    int n_kmers) {
    int k = blockIdx.x * blockDim.x + threadIdx.x;
    if (k >= n_kmers) return;
    const int*   m = resolve_map     + (size_t)k * MAX_RESOLVE;
    const float* w = resolve_weights + (size_t)k * MAX_RESOLVE;
    float acc = 0.0f;
#pragma unroll
    for (int r = 0; r < MAX_RESOLVE; ++r) {
        // unused slots have w==0 and map==0 -> contribute exactly 0
        acc = fmaf(w[r], kmer_embedding[m[r]], acc);
    }
    merged[k] = acc;
}

// ---------------------------------------------------------------------------
// Phase 2: streaming gather + exp. Bandwidth-bound (48 MB total traffic).
// The merged table is staged into LDS with CDNA5 async global->LDS b128 DMA,
// then each site does one ds_load_b32 gather + v_add + v_exp_f32.
// ---------------------------------------------------------------------------
__global__ void __launch_bounds__(BLOCK)
gather_exp_kernel(const int*   __restrict__ encoded_parent,
                  const float* __restrict__ log_site_rates,
                  const float* __restrict__ merged,
                  float*       __restrict__ out,
                  int L, int n_kmers) {
    __shared__ float tbl[TBL_MAX];

    // --- Stage merged[] into LDS via async-to-LDS b128 DMA (gfx1250 path).
    // Each b128 moves one float4. n_kmers is a multiple of 4 here (1664);
    // guard anyway by rounding up (d_ws is large, over-read is harmless).
    int nvec = (n_kmers + 3) >> 2;
    for (int c = threadIdx.x; c < nvec; c += BLOCK) {
        unsigned     lds_off = (unsigned)(size_t)(&tbl[c * 4]);   // low 32 bits = LDS byte offset
        const float* src     = merged + (size_t)c * 4;
        asm volatile("global_load_async_to_lds_b128 %0, %1, off"
                     :: "v"(lds_off), "v"(src)
                     : "memory");
    }
    // Wait for this wave's async DMA (ASYNCcnt), then make LDS visible
    // workgroup-wide.
    asm volatile("s_wait_asynccnt 0" ::: "memory");
    __syncthreads();

    // --- Streaming main loop: 4 x float4 per thread, fully coalesced b128.
    size_t blockStart = (size_t)blockIdx.x * (BLOCK * VPT);
#pragma unroll
    for (int it = 0; it < VPT / 4; ++it) {
        size_t i = blockStart + (size_t)it * (BLOCK * 4) + (size_t)threadIdx.x * 4;
        if (i + 4 <= (size_t)L) {
            int4   idx = *(const int4*)  (encoded_parent + i);
            float4 lsr = *(const float4*)(log_site_rates + i);
            float4 o;
            o.x = __expf(tbl[idx.x] + lsr.x);
            o.y = __expf(tbl[idx.y] + lsr.y);
            o.z = __expf(tbl[idx.z] + lsr.z);
            o.w = __expf(tbl[idx.w] + lsr.w);
            *(float4*)(out + i) = o;
        } else {
            for (size_t j = i; j < (size_t)L; ++j)
                out[j] = __expf(tbl[encoded_parent[j]] + log_site_rates[j]);
        }
    }
}

// ---------------------------------------------------------------------------
// Launch. Input order (setup_inputs): encoded_parent, kmer_embedding,
// log_site_rates, resolve_map, resolve_weights. Output: float32[L].
// d_ws holds merged[] (n_kmers floats).
// ---------------------------------------------------------------------------
extern "C" void kernel_launch(void* const* d_in, const int* in_sizes, int n_in,
                              void* d_out, int out_size, void* d_ws, size_t ws_size,
                              hipStream_t stream) {
    const int*   encoded_parent  = (const int*)  d_in[0];
    const float* kmer_embedding  = (const float*)d_in[1];
    const float* log_site_rates  = (const float*)d_in[2];
    const int*   resolve_map     = (const int*)  d_in[3];
    const float* resolve_weights = (const float*)d_in[4];
    float*       out             = (float*)d_out;
    float*       merged          = (float*)d_ws;

    int L       = in_sizes[0];
    int n_kmers = in_sizes[3] / MAX_RESOLVE;   // 1664 for this problem (<= TBL_MAX)

    int b1 = (n_kmers + 255) / 256;
    build_merged_kernel<<<b1, 256, 0, stream>>>(resolve_map, resolve_weights,
                                                kmer_embedding, merged, n_kmers);

    int elemsPerBlock = BLOCK * VPT;
    int b2 = (L + elemsPerBlock - 1) / elemsPerBlock;
    gather_exp_kernel<<<b2, BLOCK, 0, stream>>>(encoded_parent, log_site_rates,
                                                merged, out, L, n_kmers);
}